// GatFCM_84937273245932
// MI455X (gfx1250) — compile-verified
//
#include <hip/hip_runtime.h>

#define NN 100000
#define NE 1600000
#define DIN 480
#define DOUT 256
#define NEG_SLOPE 0.2f

typedef float v2f __attribute__((ext_vector_type(2)));
typedef float v8f __attribute__((ext_vector_type(8)));
typedef int   v4i __attribute__((ext_vector_type(4)));

#define AS1 __attribute__((address_space(1)))
#define AS3 __attribute__((address_space(3)))

// order-preserving float <-> uint encoding for atomicMax-based segment max
__device__ __forceinline__ unsigned fenc(float f) {
  unsigned u = __float_as_uint(f);
  return (u & 0x80000000u) ? ~u : (u | 0x80000000u);
}
__device__ __forceinline__ float fdec(unsigned u) {
  u = (u & 0x80000000u) ? (u & 0x7FFFFFFFu) : ~u;
  return __uint_as_float(u);
}

// ---------------------------------------------------------------------------
// init: out[n][d] = bias[d]; maxenc = 0 (== -inf in encoded order); denom = 0
// ---------------------------------------------------------------------------
__global__ __launch_bounds__(256) void gat_init(
    const float* __restrict__ bias, float* __restrict__ out,
    unsigned* __restrict__ maxenc, float* __restrict__ denom) {
  size_t i = (size_t)blockIdx.x * 256 + threadIdx.x;  // 0 .. NN*256-1
  out[i] = bias[i & 255];
  if (i < NN) { maxenc[i] = 0u; denom[i] = 0.0f; }
}

// ---------------------------------------------------------------------------
// GEMM: xl = x@Wl + bl ; xr = x@Wr + br, via V_WMMA_F32_16X16X4_F32
// grid.x = ceil(NN/128), grid.y = 8 (col tiles of 64 over [Wl | Wr] = 512 cols)
// block = 256 threads = 8 wave32; each wave computes a 16x64 strip.
// A tile staged with async global->LDS DMA; B tile staged transposed so
// B fragments are single contiguous ds_load_b64's.
// ---------------------------------------------------------------------------
__global__ __launch_bounds__(256) void gat_gemm(
    const float* __restrict__ x,
    const float* __restrict__ Wl, const float* __restrict__ bl,
    const float* __restrict__ Wr, const float* __restrict__ br,
    float* __restrict__ xl, float* __restrict__ xr) {
  __shared__ float As[128 * 36];   // [row][k]  128x32, padded stride 36
  __shared__ float BsT[64 * 36];   // [n][k]    64x32 transposed, padded stride 36

  const int tid  = threadIdx.x;
  const int lane = tid & 31;
  const int wave = tid >> 5;

  const int rowBase = blockIdx.x * 128;
  const int tileN   = blockIdx.y;               // 0..7
  const bool left   = (tileN < 4);
  const float* W    = left ? Wl : Wr;
  const float* bv   = left ? bl : br;
  float* obuf       = left ? xl : xr;
  const int colBase = (tileN & 3) * 64;         // 0..192 inside one 256-wide W

  const int m  = lane & 15;                     // A-frag row within 16
  const int kh = (lane >> 4) * 2;               // K half-select (0 or 2)
  const int n  = lane & 15;                     // B/C/D column within 16

  // B staging assignment: each thread owns 8 consecutive k's of one column
  const int bn = tid & 63;                      // column 0..63
  const int bk = (tid >> 6) * 8;                // k base 0,8,16,24

  v8f acc[4] = {};

  for (int k0 = 0; k0 < DIN; k0 += 32) {
    // ---- stage A tile 128x32 (4 x b128 async DMA per thread) ----
    // rows >= NN are clamped (their outputs are masked at the store), which
    // keeps EXEC all-ones for async issue and avoids stale LDS contents.
#pragma unroll
    for (int it = 0; it < 4; ++it) {
      int idx = tid + it * 256;
      int r = idx >> 3, kq = (idx & 7) * 4;
      int grow = rowBase + r;
      if (grow >= NN) grow = NN - 1;
      const float* gp = x + (size_t)grow * DIN + k0 + kq;
#if __has_builtin(__builtin_amdgcn_global_load_async_to_lds_b128)
      __builtin_amdgcn_global_load_async_to_lds_b128(
          (AS1 v4i*)gp, (AS3 v4i*)&As[r * 36 + kq], 0, 0);
#else
      *(float4*)&As[r * 36 + kq] = *(const float4*)gp;
#endif
    }
    // ---- stage B tile 32x64 transposed into BsT[n][k] ----
    {
      float tmp[8];
#pragma unroll
      for (int i = 0; i < 8; ++i)
        tmp[i] = W[(size_t)(k0 + bk + i) * DOUT + colBase + bn];
      *(float4*)&BsT[bn * 36 + bk]     = make_float4(tmp[0], tmp[1], tmp[2], tmp[3]);
      *(float4*)&BsT[bn * 36 + bk + 4] = make_float4(tmp[4], tmp[5], tmp[6], tmp[7]);
    }
#if __has_builtin(__builtin_amdgcn_global_load_async_to_lds_b128)
    asm volatile("s_wait_asynccnt 0x0" ::: "memory");
#endif
    __syncthreads();

#pragma unroll
    for (int kc = 0; kc < 8; ++kc) {
      const int kk = kc * 4;
      // A frag (16x4 f32): lane m, VGPR0=K(kk+kh), VGPR1=K(kk+kh+1)
      v2f a = *(const v2f*)&As[(wave * 16 + m) * 36 + kk + kh];
#pragma unroll
      for (int j = 0; j < 4; ++j) {
        // B frag (4x16 f32): contiguous b64 from transposed tile
        v2f b = *(const v2f*)&BsT[(j * 16 + n) * 36 + kk + kh];
        acc[j] = __builtin_amdgcn_wmma_f32_16x16x4_f32(
            false, a, false, b, (short)0, acc[j], false, false);
      }
    }
    __syncthreads();
  }

  // C/D layout: VGPR v, lane l -> M = v + 8*(l>=16), N = l%16
  const int rBase0 = rowBase + wave * 16 + 8 * (lane >> 4);
#pragma unroll
  for (int j = 0; j < 4; ++j) {
    int col = colBase + j * 16 + n;
    float bb = bv[col];
#pragma unroll
    for (int v = 0; v < 8; ++v) {
      int row = rBase0 + v;
      if (row < NN) obuf[(size_t)row * DOUT + col] = acc[j][v] + bb;
    }
  }
}

// ---------------------------------------------------------------------------
// edge logits: e = att . leaky_relu(xl[src] + xr[dst]); segment max via
// monotone-uint atomicMax. One wave32 per edge, 8 floats per lane.
// ---------------------------------------------------------------------------
__global__ __launch_bounds__(256) void gat_logits(
    const float* __restrict__ xl, const float* __restrict__ xr,
    const int* __restrict__ src, const int* __restrict__ dst,
    const float* __restrict__ att,
    float* __restrict__ e_out, unsigned* __restrict__ maxenc) {
  const int wave = threadIdx.x >> 5;
  const int lane = threadIdx.x & 31;
  const int edge = blockIdx.x * 8 + wave;
  if (edge >= NE) return;
  const int s = src[edge], d = dst[edge];
  const float4* pl = (const float4*)(xl + (size_t)s * DOUT);
  const float4* pr = (const float4*)(xr + (size_t)d * DOUT);
  const float4* pa = (const float4*)att;
  float sum = 0.f;
#pragma unroll
  for (int q = 0; q < 2; ++q) {
    float4 a = pl[lane * 2 + q];
    float4 b = pr[lane * 2 + q];
    float4 w = pa[lane * 2 + q];
    float h;
    h = a.x + b.x; h = h > 0.f ? h : NEG_SLOPE * h; sum += h * w.x;
    h = a.y + b.y; h = h > 0.f ? h : NEG_SLOPE * h; sum += h * w.y;
    h = a.z + b.z; h = h > 0.f ? h : NEG_SLOPE * h; sum += h * w.z;
    h = a.w + b.w; h = h > 0.f ? h : NEG_SLOPE * h; sum += h * w.w;
  }
#pragma unroll
  for (int off = 16; off > 0; off >>= 1) sum += __shfl_xor(sum, off, 32);
  if (lane == 0) {
    e_out[edge] = sum;
    atomicMax(maxenc + d, fenc(sum));
  }
}

// ---------------------------------------------------------------------------
// ex = exp(e - max[dst]); denom[dst] += ex. One thread per edge.
// ---------------------------------------------------------------------------
__global__ __launch_bounds__(256) void gat_expdenom(
    float* __restrict__ e, const int* __restrict__ dst,
    const unsigned* __restrict__ maxenc, float* __restrict__ denom) {
  int i = blockIdx.x * 256 + threadIdx.x;
  if (i >= NE) return;
  int d = dst[i];
  float mx = fdec(maxenc[d]);  // every dst here has >=1 edge -> enc > 0, finite
  float ex = __expf(e[i] - mx);
  e[i] = ex;
  atomicAdd(denom + d, ex);
}

// ---------------------------------------------------------------------------
// out[dst] += (ex/denom[dst]) * xl[src]. One wave32 per edge.
// ---------------------------------------------------------------------------
__global__ __launch_bounds__(256) void gat_scatter(
    const float* __restrict__ xl, const int* __restrict__ src,
    const int* __restrict__ dst, const float* __restrict__ ex,
    const float* __restrict__ denom, float* __restrict__ out) {
  const int wave = threadIdx.x >> 5;
  const int lane = threadIdx.x & 31;
  const int edge = blockIdx.x * 8 + wave;
  if (edge >= NE) return;
  const int s = src[edge], d = dst[edge];
  const float alpha = ex[edge] / (denom[d] + 1e-16f);
  const float4* pl = (const float4*)(xl + (size_t)s * DOUT);
  float* po = out + (size_t)d * DOUT;
#pragma unroll
  for (int q = 0; q < 2; ++q) {
    float4 v = pl[lane * 2 + q];
    int base = lane * 8 + q * 4;
    atomicAdd(po + base + 0, alpha * v.x);
    atomicAdd(po + base + 1, alpha * v.y);
    atomicAdd(po + base + 2, alpha * v.z);
    atomicAdd(po + base + 3, alpha * v.w);
  }
}

// ---------------------------------------------------------------------------
extern "C" void kernel_launch(void* const* d_in, const int* in_sizes, int n_in,
                              void* d_out, int out_size, void* d_ws, size_t ws_size,
                              hipStream_t stream) {
  const float* x    = (const float*)d_in[0];
  const int*   ei   = (const int*)d_in[1];
  const float* Wl   = (const float*)d_in[2];
  const float* bl   = (const float*)d_in[3];
  const float* Wr   = (const float*)d_in[4];
  const float* br   = (const float*)d_in[5];
  const float* att  = (const float*)d_in[6];
  const float* bias = (const float*)d_in[7];
  const int* src = ei;            // edge_index[0]
  const int* dst = ei + NE;       // edge_index[1]
  float* out = (float*)d_out;

  // workspace layout (all offsets 16B-aligned)
  char* ws = (char*)d_ws;
  float*    xl     = (float*)(ws);                      // NN*256 f32 = 102.4 MB
  float*    xr     = (float*)(ws + 102400000ull);       // NN*256 f32 = 102.4 MB
  float*    ebuf   = (float*)(ws + 204800000ull);       // NE f32     =   6.4 MB
  unsigned* maxenc = (unsigned*)(ws + 211200000ull);    // NN u32     =   0.4 MB
  float*    denom  = (float*)(ws + 211600000ull);       // NN f32     =   0.4 MB

  gat_init<<<NN, 256, 0, stream>>>(bias, out, maxenc, denom);

  dim3 ggrid((NN + 127) / 128, 8);
  gat_gemm<<<ggrid, 256, 0, stream>>>(x, Wl, bl, Wr, br, xl, xr);

  gat_logits<<<(NE + 7) / 8, 256, 0, stream>>>(xl, xr, src, dst, att, ebuf, maxenc);
  gat_expdenom<<<(NE + 255) / 256, 256, 0, stream>>>(ebuf, dst, maxenc, denom);
  gat_scatter<<<(NE + 7) / 8, 256, 0, stream>>>(xl, src, dst, ebuf, denom, out);
}